// GraphConvolution_64630667870472
// MI455X (gfx1250) — compile-verified
//
#include <hip/hip_runtime.h>

// GraphConvolution on MI455X (gfx1250):
//   phase 1: block-diagonal SpMM via wave-per-edge native global_atomic_add_f32
//   phase 2: h @ W + bias via V_WMMA_F32_16X16X4_F32 tiled through LDS

typedef __attribute__((ext_vector_type(2))) float v2f;
typedef __attribute__((ext_vector_type(8))) float v8f;

constexpr int Dk = 256;  // feature dim
constexpr int Gk = 4;    // groups
constexpr int DGk = 64;  // features per group

// Guaranteed-native fp32 atomic add (GLOBAL_ATOMIC_ADD_F32, no-return -> STOREcnt).
// L2-side fadd, round-to-nearest-even, denorms NoFlush (ISA ch.12); s_endpgm
// performs an implicit wait-idle so outstanding stores drain before wave exit.
__device__ __forceinline__ void atomic_add_f32_native(float* p, float v) {
  asm volatile("global_atomic_add_f32 %0, %1, off" : : "v"(p), "v"(v) : "memory");
}

// ---------------- zero workspace ----------------
__global__ __launch_bounds__(256) void zero_f32(float* __restrict__ p, long long n4) {
  long long i = (long long)blockIdx.x * blockDim.x + threadIdx.x;
  long long stride = (long long)gridDim.x * blockDim.x;
  float4 z = make_float4(0.f, 0.f, 0.f, 0.f);
  for (; i < n4; i += stride) ((float4*)p)[i] = z;
}

// ---------------- SpMM scatter: one wave32 per edge ----------------
// h[row, g*64 + f] += val * x[col, g*64 + f],  f = 0..63 (2 floats per lane)
__global__ __launch_bounds__(256) void spmm_scatter(
    const float* __restrict__ x, const int* __restrict__ erow,
    const int* __restrict__ ecol, const float* __restrict__ eval,
    float* __restrict__ h, int total_edges, int E) {
  int wave = (int)((blockIdx.x * blockDim.x + threadIdx.x) >> 5);
  if (wave >= total_edges) return;  // whole-wave uniform exit
  int lane = threadIdx.x & 31;
  int g = wave / E;                 // group index
  int r = erow[wave];
  int c = ecol[wave];
  float v = eval[wave];
  int fo = g * DGk + lane * 2;
  float2 xv = *(const float2*)(x + (size_t)c * Dk + fo);  // 256B/wave, coalesced
  float* dst = h + (size_t)r * Dk + fo;
  atomic_add_f32_native(dst, v * xv.x);
  atomic_add_f32_native(dst + 1, v * xv.y);
}

// ---------------- GEMM: out = h @ W + bias (fp32 WMMA) ----------------
// block tile 128x64, 8 waves of 32x32 (2x2 fragments of 16x16), Ktile=32
__global__ __launch_bounds__(256) void gemm_wmma_f32(
    const float* __restrict__ h, const float* __restrict__ W,
    const float* __restrict__ bias, float* __restrict__ out, int N) {
  __shared__ float As[128][34];  // [m][k]  pad 34: 8B-aligned pairs, conflict-free
  __shared__ float Bs[64][34];   // [n][k]  transposed: B frag = contiguous b64 like A

  const int tid = threadIdx.x;
  const int lane = tid & 31;
  const int wid = tid >> 5;
  const int half = lane >> 4;    // 0: lanes 0-15, 1: lanes 16-31
  const int mr = lane & 15;
  const int wr = (wid & 3) * 32;   // wave row offset within block tile
  const int wc = (wid >> 2) * 32;  // wave col offset within block tile
  const int row0 = blockIdx.x * 128;
  const int col0 = blockIdx.y * 64;

  v8f acc[2][2] = {};

  for (int kt = 0; kt < Dk; kt += 32) {
    // stage A: 128x32 of h (coalesced, guarded for tail rows)
#pragma unroll
    for (int i = 0; i < 16; ++i) {
      int idx = tid + i * 256;
      int rl = idx >> 5, cl = idx & 31;
      int grow = row0 + rl;
      As[rl][cl] = (grow < N) ? h[(size_t)grow * Dk + kt + cl] : 0.0f;
    }
    // stage B: 32x64 of W, stored transposed [n][k]
#pragma unroll
    for (int i = 0; i < 8; ++i) {
      int idx = tid + i * 256;
      int rl = idx >> 6, cl = idx & 63;   // rl = k-local, cl = n-local
      Bs[cl][rl] = W[(size_t)(kt + rl) * Dk + col0 + cl];
    }
    __syncthreads();

#pragma unroll
    for (int k = 0; k < 32; k += 4) {
      v2f a[2], b[2];
      // fp32 16x4 fragment layout: lanes 0-15 -> K={k,k+1}, lanes 16-31 -> K={k+2,k+3}
#pragma unroll
      for (int s = 0; s < 2; ++s) {
        a[s].x = As[wr + s * 16 + mr][k + 2 * half];
        a[s].y = As[wr + s * 16 + mr][k + 2 * half + 1];
        b[s].x = Bs[wc + s * 16 + mr][k + 2 * half];
        b[s].y = Bs[wc + s * 16 + mr][k + 2 * half + 1];
      }
#pragma unroll
      for (int sr = 0; sr < 2; ++sr)
#pragma unroll
        for (int sc = 0; sc < 2; ++sc)
          acc[sr][sc] = __builtin_amdgcn_wmma_f32_16x16x4_f32(
              false, a[sr], false, b[sc], (short)0, acc[sr][sc], false, false);
    }
    __syncthreads();
  }

  // store D + bias; C/D layout: VGPR v -> M = v + 8*half, N = mr
#pragma unroll
  for (int sr = 0; sr < 2; ++sr) {
#pragma unroll
    for (int sc = 0; sc < 2; ++sc) {
      int gc = col0 + wc + sc * 16 + mr;
      float bv = bias[gc];
#pragma unroll
      for (int v = 0; v < 8; ++v) {
        int gr = row0 + wr + sr * 16 + v + 8 * half;
        if (gr < N) out[(size_t)gr * Dk + gc] = acc[sr][sc][v] + bv;
      }
    }
  }
}

extern "C" void kernel_launch(void* const* d_in, const int* in_sizes, int n_in,
                              void* d_out, int out_size, void* d_ws, size_t ws_size,
                              hipStream_t stream) {
  const float* x    = (const float*)d_in[0];
  const int*   erow = (const int*)d_in[1];
  const int*   ecol = (const int*)d_in[2];
  const float* eval = (const float*)d_in[3];
  const float* W    = (const float*)d_in[4];
  const float* bias = (const float*)d_in[5];
  float* out = (float*)d_out;

  int N = in_sizes[0] / Dk;          // 50000
  int total_edges = in_sizes[1];     // G*E = 3.2M
  int E = total_edges / Gk;          // 800000
  float* h = (float*)d_ws;           // N*Dk fp32 accumulator

  long long n4 = (long long)N * Dk / 4;
  zero_f32<<<dim3(2048), dim3(256), 0, stream>>>(h, n4);

  int spmm_blocks = (total_edges + 7) / 8;  // 8 waves (edges) per 256-thread block
  spmm_scatter<<<dim3(spmm_blocks), dim3(256), 0, stream>>>(x, erow, ecol, eval, h,
                                                            total_edges, E);

  dim3 ggrid((N + 127) / 128, Dk / 64);
  gemm_wmma_f32<<<ggrid, dim3(256), 0, stream>>>(h, W, bias, out, N);
}